// SiamRPN_loss_27590869909512
// MI455X (gfx1250) — compile-verified
//
#include <hip/hip_runtime.h>
#include <hip/hip_bf16.h>
#include <math.h>

// SiamRPN loss for MI455X (gfx1250, wave32).
// Memory-bound masked reduction: ~141 MB read, ~0.1 GFLOP -> HBM roofline ~6us.
// Structure: 1 block per batch row, float2/float4 coalesced loads, f32 VALU math,
// wave reduction via V_WMMA_F32_16X16X4_F32 (A = per-lane value, B = ones ->
// D[m][n] = v_m + v_{m+16}, exact f32), LDS cross-wave combine, then a 1-block
// finalize kernel averaging the per-batch partials staged in d_ws.

typedef __attribute__((ext_vector_type(2))) float v2f;
typedef __attribute__((ext_vector_type(8))) float v8f;

#define LAMB_F 5.0f
#define N_ANCHORS 3125
#define BLK 256
#define NWAVES (BLK / 32)

// Exact full-wave (32-lane) f32 sum using one WMMA.
// A layout (documented): lane m<16 holds row m cols {0(K0),1(K1)}; lane m+16 holds
// row m cols {2,3}. We place v in A.V0 and 0 in A.V1, B = all ones (layout-agnostic),
// so D[m][n] = v_m + v_{m+16} for every n. D layout (documented): d[r] = row r
// (lanes 0-15) / row r+8 (lanes 16-31). Sum d[0..7] per lane, then fold halves.
__device__ __forceinline__ float wave_sum_f32(float v) {
    v2f a; a[0] = v;    a[1] = 0.0f;
    v2f b; b[0] = 1.0f; b[1] = 1.0f;
    v8f c = {};
    v8f d = __builtin_amdgcn_wmma_f32_16x16x4_f32(
        /*neg_a=*/false, a, /*neg_b=*/false, b,
        /*c_mod=*/(short)0, c, /*reuse_a=*/false, /*reuse_b=*/false);
    float t = ((d[0] + d[1]) + (d[2] + d[3])) + ((d[4] + d[5]) + (d[6] + d[7]));
    t += __shfl_xor(t, 16, 32);   // lanes 0-15 hold rows 0-7 sum, 16-31 hold rows 8-15
    return t;                     // all lanes: full 32-lane sum
}

__device__ __forceinline__ float smooth_l1(float d) {
    float a = fabsf(d);
    return (a < 1.0f) ? (0.5f * d * d) : (a - 0.5f);
}

__global__ __launch_bounds__(BLK)
void siam_batch_kernel(const float* __restrict__ c_pred,
                       const float* __restrict__ r_pred,
                       const float* __restrict__ r_target,
                       const int*   __restrict__ label,
                       float* __restrict__ part_c,
                       float* __restrict__ part_r,
                       int N) {
    const int b   = blockIdx.x;
    const int tid = threadIdx.x;

    const float2* cp = (const float2*)(c_pred   + (size_t)b * N * 2);
    const float4* rp = (const float4*)(r_pred   + (size_t)b * N * 4);
    const float4* rt = (const float4*)(r_target + (size_t)b * N * 4);
    const int*    lb = label + (size_t)b * N;

    float pc = 0.0f, nc = 0.0f;          // pos / neg counts
    float scp = 0.0f, scn = 0.0f;        // sum CE(pos), sum CE(neg)
    float sr = 0.0f;                     // sum smooth-L1 anchor-mean over pos

    for (int i = tid; i < N; i += BLK) {
        // Prefetch next tile of the dominant (regression) streams into L2.
        if (i + BLK < N) {
            __builtin_prefetch(&rp[i + BLK], 0, 0);
            __builtin_prefetch(&rt[i + BLK], 0, 0);
        }
        float2 cv = cp[i];
        int    l  = lb[i];
        float4 p  = rp[i];
        float4 t  = rt[i];

        // 2-class log-softmax via stable log-sum-exp.
        float mx  = fmaxf(cv.x, cv.y);
        float mn  = fminf(cv.x, cv.y);
        float lse = mx + log1pf(__expf(mn - mx));

        bool pos = (l == 1);
        bool neg = (l == 0);
        pc  += pos ? 1.0f : 0.0f;
        nc  += neg ? 1.0f : 0.0f;
        scp += pos ? (lse - cv.y) : 0.0f;   // -logp[...,1]
        scn += neg ? (lse - cv.x) : 0.0f;   // -logp[...,0]

        float sl = smooth_l1(p.x - t.x) + smooth_l1(p.y - t.y) +
                   smooth_l1(p.z - t.z) + smooth_l1(p.w - t.w);
        sr += pos ? (0.25f * sl) : 0.0f;
    }

    // Intra-wave reductions (uniform control flow here -> EXEC all ones for WMMA).
    pc  = wave_sum_f32(pc);
    nc  = wave_sum_f32(nc);
    scp = wave_sum_f32(scp);
    scn = wave_sum_f32(scn);
    sr  = wave_sum_f32(sr);

    __shared__ float sm[NWAVES][5];
    const int wid  = tid >> 5;
    const int lane = tid & 31;
    if (lane == 0) {
        sm[wid][0] = pc;  sm[wid][1] = nc;  sm[wid][2] = scp;
        sm[wid][3] = scn; sm[wid][4] = sr;
    }
    __syncthreads();

    if (tid == 0) {
        float PC = 0.f, NC = 0.f, SCP = 0.f, SCN = 0.f, SR = 0.f;
        for (int w = 0; w < NWAVES; ++w) {
            PC  += sm[w][0]; NC  += sm[w][1]; SCP += sm[w][2];
            SCN += sm[w][3]; SR  += sm[w][4];
        }
        float pos_mean = (PC > 0.0f) ? (SCP / PC) : 0.0f;
        float neg_mean = SCN / fmaxf(NC, 1.0f);
        part_c[b] = 0.5f * (pos_mean + neg_mean);
        part_r[b] = (PC > 0.0f) ? (SR / PC) : 0.0f;
    }
}

__global__ __launch_bounds__(BLK)
void siam_final_kernel(const float* __restrict__ part_c,
                       const float* __restrict__ part_r,
                       float* __restrict__ out,
                       int B) {
    const int tid = threadIdx.x;
    float sc = 0.0f, sr = 0.0f;
    for (int i = tid; i < B; i += BLK) {
        sc += part_c[i];
        sr += part_r[i];
    }
    sc = wave_sum_f32(sc);
    sr = wave_sum_f32(sr);

    __shared__ float sm[NWAVES][2];
    const int wid  = tid >> 5;
    const int lane = tid & 31;
    if (lane == 0) { sm[wid][0] = sc; sm[wid][1] = sr; }
    __syncthreads();

    if (tid == 0) {
        float C = 0.0f, R = 0.0f;
        for (int w = 0; w < NWAVES; ++w) { C += sm[w][0]; R += sm[w][1]; }
        float invB = 1.0f / (float)B;
        C *= invB;
        R *= invB;
        out[0] = C;                  // c_loss
        out[1] = R;                  // r_loss
        out[2] = C + LAMB_F * R;     // t_loss
    }
}

extern "C" void kernel_launch(void* const* d_in, const int* in_sizes, int n_in,
                              void* d_out, int out_size, void* d_ws, size_t ws_size,
                              hipStream_t stream) {
    const float* c_pred   = (const float*)d_in[0];
    const float* r_pred   = (const float*)d_in[1];
    const float* r_target = (const float*)d_in[2];
    const int*   label    = (const int*)d_in[3];

    const int N = N_ANCHORS;
    const int B = in_sizes[3] / N;   // label is [B, N]

    float* part_c = (float*)d_ws;    // B floats
    float* part_r = part_c + B;      // B floats  (8 KB total, well under ws_size)

    siam_batch_kernel<<<B, BLK, 0, stream>>>(c_pred, r_pred, r_target, label,
                                             part_c, part_r, N);
    siam_final_kernel<<<1, BLK, 0, stream>>>(part_c, part_r, (float*)d_out, B);
}